// Net_14267881357677
// MI455X (gfx1250) — compile-verified
//
#include <hip/hip_runtime.h>
#include <math.h>

typedef float v2f __attribute__((ext_vector_type(2)));
typedef float v8f __attribute__((ext_vector_type(8)));

#define WAVES_PER_BLOCK 8
#define TPB 256

__device__ __forceinline__ v8f wmma4(v2f a, v2f b, v8f c) {
  // V_WMMA_F32_16X16X4_F32 : D = A(16x4,f32) * B(4x16,f32) + C(16x16,f32)
  return __builtin_amdgcn_wmma_f32_16x16x4_f32(false, a, false, b, (short)0, c,
                                               false, false);
}

// ---------------- edge scatter kernels (segment_sum) ----------------

// layer 1: features are scalar (NUM_FEATURES = 1)
__global__ void scatter1_kernel(const float* __restrict__ x,
                                const long long* __restrict__ ei,
                                float* __restrict__ agg1, int nEdges) {
  int e = blockIdx.x * blockDim.x + threadIdx.x;
  if (e >= nEdges) return;
  long long s = ei[e];
  long long d = ei[(size_t)nEdges + e];
  unsafeAtomicAdd(&agg1[d], x[s]);
}

// layers 2..5: 32-wide rows; 8 lanes per edge, float4 gather + 4 f32 atomics
__global__ void scatter32_kernel(const float* __restrict__ h,
                                 const long long* __restrict__ ei,
                                 float* __restrict__ agg, int nEdges) {
  long long t = (long long)blockIdx.x * blockDim.x + threadIdx.x;
  long long e = t >> 3;
  if (e >= nEdges) return;
  int q = ((int)t & 7) * 4;
  long long s = ei[e];
  long long d = ei[(size_t)nEdges + e];
  const float4 v = *(const float4*)(h + (size_t)s * 32 + q);
  float* dp = agg + (size_t)d * 32 + q;
  unsafeAtomicAdd(dp + 0, v.x);
  unsafeAtomicAdd(dp + 1, v.y);
  unsafeAtomicAdd(dp + 2, v.z);
  unsafeAtomicAdd(dp + 3, v.w);
}

// ---------------- GIN layer 1 (scalar input -> WMMA second GEMM) ------------

__global__ void gin1_kernel(const float* __restrict__ x,
                            const float* __restrict__ agg1,
                            const float* __restrict__ w1,   // (1,32)
                            const float* __restrict__ b1,   // (32)
                            const float* __restrict__ w2,   // (32,32)
                            const float* __restrict__ b2,   // (32)
                            float* __restrict__ pre, float* __restrict__ stats,
                            int n) {
  __shared__ float ssum[32], ssq[32];
  if (threadIdx.x < 32) { ssum[threadIdx.x] = 0.f; ssq[threadIdx.x] = 0.f; }
  __syncthreads();

  const int lane = threadIdx.x & 31;
  const int l = lane & 15;
  const int half = lane >> 4;
  const int wave = threadIdx.x >> 5;
  const int base = (blockIdx.x * WAVES_PER_BLOCK + wave) * 16;

  float s0 = 0.f, q0 = 0.f, s1 = 0.f, q1 = 0.f;
  if (base < n) {
    const float z = x[base + l] + agg1[base + l];
    v8f acc0 = {}; v8f acc1 = {};
#pragma unroll
    for (int kk = 0; kk < 8; ++kk) {
      const int k0 = kk * 4 + half * 2;
      v2f a;  // hidden activation generated directly in A layout
      a.x = fmaxf(fmaf(z, w1[k0],     b1[k0]),     0.f);
      a.y = fmaxf(fmaf(z, w1[k0 + 1], b1[k0 + 1]), 0.f);
      v2f bb0, bb1;
      bb0.x = w2[k0 * 32 + l];       bb0.y = w2[(k0 + 1) * 32 + l];
      bb1.x = w2[k0 * 32 + 16 + l];  bb1.y = w2[(k0 + 1) * 32 + 16 + l];
      acc0 = wmma4(a, bb0, acc0);
      acc1 = wmma4(a, bb1, acc1);
    }
    const float bias0 = b2[l], bias1 = b2[16 + l];
#pragma unroll
    for (int r = 0; r < 8; ++r) {
      const int row = base + r + half * 8;
      const float v0 = fmaxf(acc0[r] + bias0, 0.f);  // outer ReLU
      const float v1 = fmaxf(acc1[r] + bias1, 0.f);
      pre[(size_t)row * 32 + l] = v0;
      pre[(size_t)row * 32 + 16 + l] = v1;
      s0 += v0; q0 += v0 * v0; s1 += v1; q1 += v1 * v1;
    }
  }
  atomicAdd(&ssum[l], s0);      atomicAdd(&ssq[l], q0);
  atomicAdd(&ssum[16 + l], s1); atomicAdd(&ssq[16 + l], q1);
  __syncthreads();
  if (threadIdx.x < 32)      unsafeAtomicAdd(&stats[threadIdx.x], ssum[threadIdx.x]);
  else if (threadIdx.x < 64) unsafeAtomicAdd(&stats[threadIdx.x], ssq[threadIdx.x - 32]);
}

// ---------------- GIN layers 2..5: z=(h+agg); relu(z@w1+b1)@w2+b2 ----------

__global__ void gin_mlp_kernel(const float* __restrict__ h,
                               const float* __restrict__ agg,
                               const float* __restrict__ w1,
                               const float* __restrict__ b1v,
                               const float* __restrict__ w2,
                               const float* __restrict__ b2v,
                               float* __restrict__ pre,
                               float* __restrict__ stats, int n) {
  __shared__ float ssum[32], ssq[32];
  __shared__ float amem[WAVES_PER_BLOCK][16][33];  // padded vs bank conflicts
  if (threadIdx.x < 32) { ssum[threadIdx.x] = 0.f; ssq[threadIdx.x] = 0.f; }
  __syncthreads();

  const int lane = threadIdx.x & 31;
  const int l = lane & 15;
  const int half = lane >> 4;
  const int wave = threadIdx.x >> 5;
  const int base = (blockIdx.x * WAVES_PER_BLOCK + wave) * 16;

  float s0 = 0.f, q0 = 0.f, s1 = 0.f, q1 = 0.f;
  if (base < n) {
    const float* hrow = h + (size_t)(base + l) * 32;
    const float* arow = agg + (size_t)(base + l) * 32;
    // ---- GEMM1: (h+agg) @ w1 ----
    v8f acc0 = {}; v8f acc1 = {};
#pragma unroll
    for (int kk = 0; kk < 8; ++kk) {
      const int k0 = kk * 4 + half * 2;
      const float2 hz = *(const float2*)(hrow + k0);
      const float2 az = *(const float2*)(arow + k0);
      v2f a; a.x = hz.x + az.x; a.y = hz.y + az.y;
      v2f bb0, bb1;
      bb0.x = w1[k0 * 32 + l];       bb0.y = w1[(k0 + 1) * 32 + l];
      bb1.x = w1[k0 * 32 + 16 + l];  bb1.y = w1[(k0 + 1) * 32 + 16 + l];
      acc0 = wmma4(a, bb0, acc0);
      acc1 = wmma4(a, bb1, acc1);
    }
    // hidden ReLU, stage through LDS to convert C-layout -> A-layout
    const float hb0 = b1v[l], hb1 = b1v[16 + l];
#pragma unroll
    for (int r = 0; r < 8; ++r) {
      const int rr = r + half * 8;
      amem[wave][rr][l]      = fmaxf(acc0[r] + hb0, 0.f);
      amem[wave][rr][16 + l] = fmaxf(acc1[r] + hb1, 0.f);
    }
    __builtin_amdgcn_wave_barrier();  // wave-synchronous LDS RAW
    // ---- GEMM2: a @ w2 ----
    v8f o0 = {}; v8f o1 = {};
#pragma unroll
    for (int kk = 0; kk < 8; ++kk) {
      const int k0 = kk * 4 + half * 2;
      v2f a; a.x = amem[wave][l][k0]; a.y = amem[wave][l][k0 + 1];
      v2f bb0, bb1;
      bb0.x = w2[k0 * 32 + l];       bb0.y = w2[(k0 + 1) * 32 + l];
      bb1.x = w2[k0 * 32 + 16 + l];  bb1.y = w2[(k0 + 1) * 32 + 16 + l];
      o0 = wmma4(a, bb0, o0);
      o1 = wmma4(a, bb1, o1);
    }
    const float bias0 = b2v[l], bias1 = b2v[16 + l];
#pragma unroll
    for (int r = 0; r < 8; ++r) {
      const int row = base + r + half * 8;
      const float v0 = fmaxf(o0[r] + bias0, 0.f);  // outer ReLU
      const float v1 = fmaxf(o1[r] + bias1, 0.f);
      pre[(size_t)row * 32 + l] = v0;
      pre[(size_t)row * 32 + 16 + l] = v1;
      s0 += v0; q0 += v0 * v0; s1 += v1; q1 += v1 * v1;
    }
  }
  atomicAdd(&ssum[l], s0);      atomicAdd(&ssq[l], q0);
  atomicAdd(&ssum[16 + l], s1); atomicAdd(&ssq[16 + l], q1);
  __syncthreads();
  if (threadIdx.x < 32)      unsafeAtomicAdd(&stats[threadIdx.x], ssum[threadIdx.x]);
  else if (threadIdx.x < 64) unsafeAtomicAdd(&stats[threadIdx.x], ssq[threadIdx.x - 32]);
}

// ---------------- BatchNorm apply (batch stats, biased variance) ------------

__global__ void bn_apply_kernel(const float* __restrict__ pre,
                                float* __restrict__ out,
                                const float* __restrict__ stats,
                                const float* __restrict__ gamma,
                                const float* __restrict__ beta, int n) {
  long long i = (long long)blockIdx.x * blockDim.x + threadIdx.x;
  if (i >= (long long)n * 32) return;
  const int c = (int)(i & 31);
  const float inv_n = 1.0f / (float)n;
  const float mu = stats[c] * inv_n;
  const float var = stats[32 + c] * inv_n - mu * mu;
  const float scale = gamma[c] * rsqrtf(var + 1e-5f);
  const float shift = beta[c] - mu * scale;
  out[i] = pre[i] * scale + shift;
}

// ---------------- head: relu(h@fc1+b) @ fc2 + b -> log_softmax --------------

__global__ void head_kernel(const float* __restrict__ h,
                            const float* __restrict__ fw1,
                            const float* __restrict__ fb1,
                            const float* __restrict__ fw2,
                            const float* __restrict__ fb2,
                            float* __restrict__ out, int n) {
  __shared__ float amem[WAVES_PER_BLOCK][16][33];
  const int lane = threadIdx.x & 31;
  const int l = lane & 15;
  const int half = lane >> 4;
  const int wave = threadIdx.x >> 5;
  const int base = (blockIdx.x * WAVES_PER_BLOCK + wave) * 16;
  if (base >= n) return;  // wave-uniform

  const float* hrow = h + (size_t)(base + l) * 32;
  v8f acc0 = {}; v8f acc1 = {};
#pragma unroll
  for (int kk = 0; kk < 8; ++kk) {
    const int k0 = kk * 4 + half * 2;
    const float2 hz = *(const float2*)(hrow + k0);
    v2f a; a.x = hz.x; a.y = hz.y;
    v2f bb0, bb1;
    bb0.x = fw1[k0 * 32 + l];       bb0.y = fw1[(k0 + 1) * 32 + l];
    bb1.x = fw1[k0 * 32 + 16 + l];  bb1.y = fw1[(k0 + 1) * 32 + 16 + l];
    acc0 = wmma4(a, bb0, acc0);
    acc1 = wmma4(a, bb1, acc1);
  }
  const float bias0 = fb1[l], bias1 = fb1[16 + l];
#pragma unroll
  for (int r = 0; r < 8; ++r) {
    const int rr = r + half * 8;
    amem[wave][rr][l]      = fmaxf(acc0[r] + bias0, 0.f);
    amem[wave][rr][16 + l] = fmaxf(acc1[r] + bias1, 0.f);
  }
  __builtin_amdgcn_wave_barrier();
  if (half == 0) {
    const int node = base + l;
    float l0 = fb2[0], l1 = fb2[1];
#pragma unroll
    for (int c = 0; c < 32; ++c) {
      const float av = amem[wave][l][c];
      l0 = fmaf(av, fw2[c * 2 + 0], l0);
      l1 = fmaf(av, fw2[c * 2 + 1], l1);
    }
    const float m = fmaxf(l0, l1);
    const float lse = m + logf(expf(l0 - m) + expf(l1 - m));
    out[(size_t)node * 2 + 0] = l0 - lse;
    out[(size_t)node * 2 + 1] = l1 - lse;
  }
}

// ---------------- host orchestration ----------------------------------------

extern "C" void kernel_launch(void* const* d_in, const int* in_sizes, int n_in,
                              void* d_out, int out_size, void* d_ws,
                              size_t ws_size, hipStream_t stream) {
  const float* x       = (const float*)d_in[0];
  const long long* ei  = (const long long*)d_in[1];  // int64 edge_index (2,E)
  const float* c1w1    = (const float*)d_in[2];
  const float* c1b1    = (const float*)d_in[3];
  const float* c1w2    = (const float*)d_in[4];
  const float* c1b2    = (const float*)d_in[5];
  const float* cw1     = (const float*)d_in[6];
  const float* cb1     = (const float*)d_in[7];
  const float* cw2     = (const float*)d_in[8];
  const float* cb2     = (const float*)d_in[9];
  const float* gamma   = (const float*)d_in[10];
  const float* beta    = (const float*)d_in[11];
  const float* fw1     = (const float*)d_in[12];
  const float* fb1     = (const float*)d_in[13];
  const float* fw2     = (const float*)d_in[14];
  const float* fb2     = (const float*)d_in[15];
  float* outp = (float*)d_out;

  const int N = in_sizes[0];       // x is (N, 1)
  const int E = in_sizes[1] / 2;   // edge_index is (2, E)

  float* hA    = (float*)d_ws;          // pre-BN scratch [N,32]
  float* hB    = hA + (size_t)N * 32;   // current activations [N,32]
  float* agg   = hB + (size_t)N * 32;   // segment_sum target [N,32]
  float* agg1  = agg + (size_t)N * 32;  // scalar segment_sum [N]
  float* stats = agg1 + N;              // [64]: sums[32], sumsq[32]

  const int nwaves = (N + 15) / 16;
  const int nblocks = (nwaves + WAVES_PER_BLOCK - 1) / WAVES_PER_BLOCK;
  const unsigned bnBlocks = (unsigned)(((long long)N * 32 + 255) / 256);
  const unsigned scBlocks = (unsigned)(((long long)E * 8 + 255) / 256);

  // ---- layer 1 (scalar features) ----
  hipMemsetAsync(agg1, 0, (size_t)N * sizeof(float), stream);
  hipMemsetAsync(stats, 0, 64 * sizeof(float), stream);
  scatter1_kernel<<<(E + 255) / 256, 256, 0, stream>>>(x, ei, agg1, E);
  gin1_kernel<<<nblocks, TPB, 0, stream>>>(x, agg1, c1w1, c1b1, c1w2, c1b2,
                                           hA, stats, N);
  bn_apply_kernel<<<bnBlocks, 256, 0, stream>>>(hA, hB, stats, gamma, beta, N);

  // ---- layers 2..5 ----
  for (int i = 0; i < 4; ++i) {
    hipMemsetAsync(agg, 0, (size_t)N * 32 * sizeof(float), stream);
    hipMemsetAsync(stats, 0, 64 * sizeof(float), stream);
    scatter32_kernel<<<scBlocks, 256, 0, stream>>>(hB, ei, agg, E);
    gin_mlp_kernel<<<nblocks, TPB, 0, stream>>>(
        hB, agg, cw1 + (size_t)i * 32 * 32, cb1 + (size_t)i * 32,
        cw2 + (size_t)i * 32 * 32, cb2 + (size_t)i * 32, hA, stats, N);
    bn_apply_kernel<<<bnBlocks, 256, 0, stream>>>(
        hA, hB, stats, gamma + (size_t)(i + 1) * 32, beta + (size_t)(i + 1) * 32, N);
  }

  // ---- head ----
  head_kernel<<<nblocks, TPB, 0, stream>>>(hB, fw1, fb1, fw2, fb2, outp, N);
}